// MultiGraphConvolution_Layer_28656021799304
// MI455X (gfx1250) — compile-verified
//
#include <hip/hip_runtime.h>

#define F_IN  128
#define F_OUT 64
#define SLOPE 0.2f

typedef __attribute__((ext_vector_type(2))) float v2f;
typedef __attribute__((ext_vector_type(8))) float v8f;

__device__ __forceinline__ float leaky(float e) { return e > 0.f ? e : SLOPE * e; }

// Sign-aware float atomic max via integer atomics (monotone-correct for all sign mixes).
__device__ __forceinline__ void atomicMaxF32(float* addr, float v) {
  if (v >= 0.f) atomicMax((int*)addr, __float_as_int(v));
  else          atomicMin((unsigned int*)addr, (unsigned int)__float_as_int(v));
}

// ---------------------------------------------------------------------------
// Kernel 1: xp = x @ W  via V_WMMA_F32_16X16X4_F32.
// One wave32 computes a 16-row x 64-col strip: 4 C tiles, K loop of 32 steps.
//   A frag (16x4 f32): lane l holds x[row0+(l&15)][4*ks + 2*(l>>4) + {0,1}]
//                      -> one global_load_b64 per k-step (row index clamped,
//                         no per-lane EXEC masking; OOB rows are never stored)
//   B frag (4x16 f32): lane l holds {W[k][n], W[k+1][n]}, n = nt*16+(l&15).
//                      W is staged in LDS as K-pair-interleaved float2 so each
//                      fragment is one contiguous ds_load_b64.
//   C tile (16x16 f32): reg r, lane l -> row r + 8*(l>>4), col (l&15)
// ---------------------------------------------------------------------------
__global__ __launch_bounds__(256) void k_gemm(const float* __restrict__ x,
                                              const float* __restrict__ W,
                                              float* __restrict__ xp, int N) {
  // sW2[kk*64 + n] = { W[2*kk][n], W[2*kk+1][n] },  kk in [0,64), n in [0,64)
  __shared__ float2 sW2[(F_IN / 2) * F_OUT];
  for (int t = threadIdx.x; t < (F_IN / 2) * F_OUT; t += 256) {
    const int kk = t >> 6;
    const int n  = t & 63;
    float2 p;
    p.x = W[(2 * kk)     * F_OUT + n];
    p.y = W[(2 * kk + 1) * F_OUT + n];
    sW2[t] = p;
  }
  __syncthreads();

  const int wave = threadIdx.x >> 5;
  const int lane = threadIdx.x & 31;
  const int row0 = (blockIdx.x * 8 + wave) * 16;
  if (row0 >= N) return;                 // wave-uniform: EXEC stays all-1s for WMMA

  const int m  = lane & 15;
  const int hb = lane >> 4;              // which half of the wave
  int row = row0 + m;
  if (row > N - 1) row = N - 1;          // clamp: avoids EXEC masking in the loop
  const float* xrow = x + (size_t)row * F_IN + 2 * hb;
  const float2* wb  = sW2 + (size_t)hb * F_OUT + m;  // pair index kk = 2*ks + hb

  v8f c0 = {}, c1 = {}, c2 = {}, c3 = {};
#pragma unroll 8
  for (int ks = 0; ks < F_IN / 4; ++ks) {
    const float2 at = *(const float2*)(xrow + 4 * ks);          // global_load_b64
    v2f a;  a.x = at.x;  a.y = at.y;
    const float2* wk = wb + (size_t)(2 * ks) * F_OUT;
    const float2 t0 = wk[0];                                    // ds_load_b64 x4
    const float2 t1 = wk[16];
    const float2 t2 = wk[32];
    const float2 t3 = wk[48];
    v2f b0, b1, b2, b3;
    b0.x = t0.x; b0.y = t0.y;
    b1.x = t1.x; b1.y = t1.y;
    b2.x = t2.x; b2.y = t2.y;
    b3.x = t3.x; b3.y = t3.y;
    c0 = __builtin_amdgcn_wmma_f32_16x16x4_f32(false, a, false, b0, (short)0, c0, false, false);
    c1 = __builtin_amdgcn_wmma_f32_16x16x4_f32(false, a, false, b1, (short)0, c1, false, false);
    c2 = __builtin_amdgcn_wmma_f32_16x16x4_f32(false, a, false, b2, (short)0, c2, false, false);
    c3 = __builtin_amdgcn_wmma_f32_16x16x4_f32(false, a, false, b3, (short)0, c3, false, false);
  }

#pragma unroll
  for (int r = 0; r < 8; ++r) {
    const int rr = row0 + r + 8 * hb;
    if (rr < N) {
      float* o = xp + (size_t)rr * F_OUT + m;
      o[0]  = c0[r];
      o[16] = c1[r];
      o[32] = c2[r];
      o[48] = c3[r];
    }
  }
}

// ---------------------------------------------------------------------------
// Kernel 2: per-node attention logits. One wave32 per node; float2 per lane,
// wave shuffle reduction (warpSize == 32 on gfx1250).
// ---------------------------------------------------------------------------
__global__ __launch_bounds__(256) void k_alpha(const float* __restrict__ xp,
                                               const float* __restrict__ a_src,
                                               const float* __restrict__ a_dst,
                                               float* __restrict__ as,
                                               float* __restrict__ ad, int N) {
  const int wave = threadIdx.x >> 5;
  const int lane = threadIdx.x & 31;
  const int i = blockIdx.x * 8 + wave;
  if (i >= N) return;
  const float2 xv = *(const float2*)(xp + (size_t)i * F_OUT + 2 * lane);
  const float2 sv = *(const float2*)(a_src + 2 * lane);
  const float2 dv = *(const float2*)(a_dst + 2 * lane);
  float s = xv.x * sv.x + xv.y * sv.y;
  float d = xv.x * dv.x + xv.y * dv.y;
  for (int off = 16; off > 0; off >>= 1) {
    s += __shfl_down(s, off);
    d += __shfl_down(d, off);
  }
  if (lane == 0) { as[i] = s; ad[i] = d; }
}

// ---------------------------------------------------------------------------
// Kernel 3: init — zero accumulator (d_out), seed emax with self-loop logit,
// zero edge-only denom.
// ---------------------------------------------------------------------------
__global__ __launch_bounds__(256) void k_init(const float* __restrict__ as,
                                              const float* __restrict__ ad,
                                              float* __restrict__ emax,
                                              float* __restrict__ denom,
                                              float* __restrict__ out, int N) {
  const long long idx = (long long)blockIdx.x * 256 + threadIdx.x;
  if (idx < (long long)N * F_OUT) out[idx] = 0.f;
  if (idx < (long long)N) {
    emax[idx]  = leaky(as[idx] + ad[idx]);
    denom[idx] = 0.f;
  }
}

// Kernel 4: per-edge segment max.
__global__ __launch_bounds__(256) void k_edge_max(const int* __restrict__ ei,
                                                  const float* __restrict__ as,
                                                  const float* __restrict__ ad,
                                                  float* __restrict__ emax, int E) {
  const int e = blockIdx.x * 256 + threadIdx.x;
  if (e >= E) return;
  const int s = ei[e];
  const int d = ei[E + e];
  atomicMaxF32(emax + d, leaky(as[s] + ad[d]));
}

// Kernel 5: per-edge unnormalized softmax weight + denom accumulation.
__global__ __launch_bounds__(256) void k_edge_w(const int* __restrict__ ei,
                                                const float* __restrict__ as,
                                                const float* __restrict__ ad,
                                                const float* __restrict__ emax,
                                                float* __restrict__ wbuf,
                                                float* __restrict__ denom, int E) {
  const int e = blockIdx.x * 256 + threadIdx.x;
  if (e >= E) return;
  const int s = ei[e];
  const int d = ei[E + e];
  const float w = __expf(leaky(as[s] + ad[d]) - emax[d]);
  wbuf[e] = w;
  atomicAdd(denom + d, w);
}

// ---------------------------------------------------------------------------
// Kernel 6: scatter-aggregate. One wave32 per edge; lane handles a float2 of
// the 64-wide feature row -> fully coalesced 256B gather + 64 f32 atomics
// into the (L2-resident) accumulator.
// ---------------------------------------------------------------------------
__global__ __launch_bounds__(256) void k_agg(const int* __restrict__ ei,
                                             const float* __restrict__ wbuf,
                                             const float* __restrict__ xp,
                                             float* __restrict__ out, int E) {
  const int wave = threadIdx.x >> 5;
  const int lane = threadIdx.x & 31;
  const int e = blockIdx.x * 8 + wave;
  if (e >= E) return;
  const int s = ei[e];
  const int d = ei[E + e];
  const float w = wbuf[e];
  const float2 xv = *(const float2*)(xp + (size_t)s * F_OUT + 2 * lane);
  float* o = out + (size_t)d * F_OUT + 2 * lane;
  atomicAdd(o,     w * xv.x);
  atomicAdd(o + 1, w * xv.y);
}

// ---------------------------------------------------------------------------
// Kernel 7: finalize — add self-loop term, normalize by (denom + w_self),
// add bias, ReLU. One wave32 per node, in place on d_out.
// ---------------------------------------------------------------------------
__global__ __launch_bounds__(256) void k_final(const float* __restrict__ xp,
                                               const float* __restrict__ as,
                                               const float* __restrict__ ad,
                                               const float* __restrict__ emax,
                                               const float* __restrict__ denom,
                                               const float* __restrict__ bias,
                                               float* __restrict__ out, int N) {
  const int wave = threadIdx.x >> 5;
  const int lane = threadIdx.x & 31;
  const int i = blockIdx.x * 8 + wave;
  if (i >= N) return;
  const float eself = leaky(as[i] + ad[i]);
  const float wself = __expf(eself - emax[i]);
  const float inv   = 1.f / (denom[i] + wself);
  float* o = out + (size_t)i * F_OUT + 2 * lane;
  const float2 xv = *(const float2*)(xp + (size_t)i * F_OUT + 2 * lane);
  const float2 bv = *(const float2*)(bias + 2 * lane);
  const float r0 = (o[0] + wself * xv.x) * inv + bv.x;
  const float r1 = (o[1] + wself * xv.y) * inv + bv.y;
  o[0] = fmaxf(r0, 0.f);
  o[1] = fmaxf(r1, 0.f);
}

extern "C" void kernel_launch(void* const* d_in, const int* in_sizes, int n_in,
                              void* d_out, int out_size, void* d_ws, size_t ws_size,
                              hipStream_t stream) {
  const float* x     = (const float*)d_in[0];
  const float* W     = (const float*)d_in[1];
  const float* a_src = (const float*)d_in[2];
  const float* a_dst = (const float*)d_in[3];
  const float* bias  = (const float*)d_in[4];
  const int*   ei    = (const int*)d_in[5];
  const int N = in_sizes[0] / F_IN;
  const int E = in_sizes[5] / 2;

  float* ws    = (float*)d_ws;
  float* xp    = ws;                       // N*64
  float* as    = xp + (size_t)N * F_OUT;   // N
  float* ad    = as + N;                   // N
  float* emax  = ad + N;                   // N
  float* denom = emax + N;                 // N
  float* wbuf  = denom + N;                // E
  float* out   = (float*)d_out;            // N*64, also the atomic accumulator

  const int tiles = (N + 15) / 16;
  k_gemm<<<(tiles + 7) / 8, 256, 0, stream>>>(x, W, xp, N);
  k_alpha<<<(N + 7) / 8, 256, 0, stream>>>(xp, a_src, a_dst, as, ad, N);
  const long long tot = (long long)N * F_OUT;
  k_init<<<(int)((tot + 255) / 256), 256, 0, stream>>>(as, ad, emax, denom, out, N);
  k_edge_max<<<(E + 255) / 256, 256, 0, stream>>>(ei, as, ad, emax, E);
  k_edge_w<<<(E + 255) / 256, 256, 0, stream>>>(ei, as, ad, emax, wbuf, denom, E);
  k_agg<<<(E + 7) / 8, 256, 0, stream>>>(ei, wbuf, xp, out, E);
  k_final<<<(N + 7) / 8, 256, 0, stream>>>(xp, as, ad, emax, denom, bias, out, N);
}